// CutMix_73641509257715
// MI455X (gfx1250) — compile-verified
//
#include <hip/hip_runtime.h>

// ---------------------------------------------------------------------------
// CutMix for MI455X (gfx1250): pure streaming-memory op.
//   x : (128, 3, 384, 384) f32   -> x2 (same shape)
//   labels : (128, 1000) f32     -> mixed_labels
// Roofline: ~0.57 GB moved, 0 FLOPs -> HBM-bound (~24 us @ 23.3 TB/s).
// Strategy: wave32 + b128 vector memory + NT stores + global_prefetch_b8.
// ---------------------------------------------------------------------------

typedef float v4f __attribute__((ext_vector_type(4)));

#define BD   128
#define CD   3
#define HD   384
#define WD   384
#define NCLS 1000
#define W4   (WD / 4)                       // 96 float4 per row
#define ROW4_PER_IMG (CD * HD * W4)         // 110592 float4 per image
#define TOTAL4 (BD * ROW4_PER_IMG)          // 14,155,776 float4 total
#define LBL_TOTAL (BD * NCLS)

struct BoxParams { int start; int end; int mixb; int pad; };

// --- tiny per-batch precompute: cut window + mix index ----------------------
__global__ void cutmix_params_kernel(const float* __restrict__ lambdaVal,
                                     const float* __restrict__ start_vals,
                                     const int*   __restrict__ mix_idx,
                                     BoxParams*   __restrict__ ws) {
    int b = threadIdx.x;
    if (b < BD) {
        float lam  = lambdaVal[b];
        int   cut  = (int)((float)HD * sqrtf(1.0f - lam));         // trunc, as jnp astype(int32)
        int   st   = (int)(start_vals[b] * (float)(HD - cut));     // trunc
        BoxParams p;
        p.start = st;
        p.end   = st + cut;
        p.mixb  = mix_idx[b];
        p.pad   = 0;
        ws[b] = p;
    }
}

// --- main image kernel: one float4 per lane per iteration -------------------
__global__ __launch_bounds__(256)
void cutmix_image_kernel(const v4f* __restrict__ x4,
                         const BoxParams* __restrict__ pp,
                         v4f* __restrict__ out4) {
    const int stride = gridDim.x * blockDim.x;
    for (int i = blockIdx.x * blockDim.x + threadIdx.x; i < TOTAL4; i += stride) {
        // Software prefetch the next grid-stride line (global_prefetch_b8):
        // hides HBM latency behind this iteration's mask arithmetic.
        if (i + stride < TOTAL4)
            __builtin_prefetch((const void*)(x4 + i + stride), 0, 1);

        // Decode (b, h, w4) from flat float4 index. c is irrelevant to the mask.
        int w4 = i % W4;
        int t  = i / W4;          // row index over (b, c, h)
        int h  = t % HD;
        int b  = t / (HD * CD);

        BoxParams p = pp[b];      // near-uniform per wave; hits L1/L2

        v4f v = x4[i];            // global_load_b128

        // Row inside the cut window for this batch?
        if (h >= p.start && h < p.end) {
            int w0 = w4 * 4;
            // Only touch the mix source if this lane's 4-column span
            // intersects [start, end) -- skips ~half the candidate traffic.
            if (w0 < p.end && (w0 + 4) > p.start) {
                int mi = i + (p.mixb - b) * ROW4_PER_IMG;   // same (c,h,w), batch=mixb
                v4f m = x4[mi];                             // global_load_b128
                if (w0 + 0 >= p.start && w0 + 0 < p.end) v.x = m.x;
                if (w0 + 1 >= p.start && w0 + 1 < p.end) v.y = m.y;
                if (w0 + 2 >= p.start && w0 + 2 < p.end) v.z = m.z;
                if (w0 + 3 >= p.start && w0 + 3 < p.end) v.w = m.w;
            }
        }

        // Output is write-once / never re-read: non-temporal b128 store keeps
        // the 192MB L2 free for the 1.5x-reused input stream.
        __builtin_nontemporal_store(v, out4 + i);
    }
}

// --- label lerp: trivial ----------------------------------------------------
__global__ __launch_bounds__(256)
void cutmix_labels_kernel(const float* __restrict__ labels,
                          const float* __restrict__ lambdaVal,
                          const int*   __restrict__ mix_idx,
                          float*       __restrict__ out) {
    int i = blockIdx.x * blockDim.x + threadIdx.x;
    if (i >= LBL_TOTAL) return;
    int b = i / NCLS;
    int k = i - b * NCLS;
    float lam = lambdaVal[b];
    float a   = labels[i];
    float c   = labels[mix_idx[b] * NCLS + k];
    float r   = lam * a + (1.0f - lam) * c;
    __builtin_nontemporal_store(r, out + i);
}

extern "C" void kernel_launch(void* const* d_in, const int* in_sizes, int n_in,
                              void* d_out, int out_size, void* d_ws, size_t ws_size,
                              hipStream_t stream) {
    const float* x          = (const float*)d_in[0];
    const float* labels     = (const float*)d_in[1];
    const float* lambdaVal  = (const float*)d_in[2];
    const float* start_vals = (const float*)d_in[3];
    const int*   mix_idx    = (const int*)  d_in[4];

    float* out_x2     = (float*)d_out;
    float* out_labels = out_x2 + (size_t)BD * CD * HD * WD;

    BoxParams* params = (BoxParams*)d_ws;   // 128 * 16 B scratch

    // 1) per-batch window params (one tiny block)
    cutmix_params_kernel<<<1, 128, 0, stream>>>(lambdaVal, start_vals, mix_idx, params);

    // 2) image blend: 4 float4 per thread via grid-stride (exactly covers TOTAL4)
    const int threads = 256;
    const int iters   = 4;
    const int blocks  = (TOTAL4 + threads * iters - 1) / (threads * iters);  // 13824
    cutmix_image_kernel<<<blocks, threads, 0, stream>>>(
        (const v4f*)x, params, (v4f*)out_x2);

    // 3) labels
    cutmix_labels_kernel<<<(LBL_TOTAL + 255) / 256, 256, 0, stream>>>(
        labels, lambdaVal, mix_idx, out_labels);
}